// Embedding_LS_64802466562285
// MI455X (gfx1250) — compile-verified
//
#include <hip/hip_runtime.h>

// ---------------------------------------------------------------------------
// Embedding gather: out[b, e] = W[e, y[b]],  W: [512, 128000] f32 row-major.
// Pure bandwidth problem (~288 MB total traffic -> ~12.4 us at 23.3 TB/s).
// gfx1250 path: async gather (global -> LDS, ASYNCcnt) + async coalesced
// store (LDS -> global, b128), one wave32 per output row.
// Address math restructured so the per-lane offset is a single 32-bit VGPR
// value (lane*vocab*4 < 2^31) and the per-iteration stride is a uniform
// scalar pointer increment -> GVS (saddr + vaddr) addressing, minimal VALU.
// ---------------------------------------------------------------------------

#define AS_LDS    __attribute__((address_space(3)))
#define AS_GLOBAL __attribute__((address_space(1)))

typedef int v4i __attribute__((ext_vector_type(4)));

#if __has_builtin(__builtin_amdgcn_global_load_async_to_lds_b32) && \
    __has_builtin(__builtin_amdgcn_global_store_async_from_lds_b128)
#define USE_ASYNC_LDS 1
#else
#define USE_ASYNC_LDS 0
#endif

// generic LDS pointer's low 32 bits are the LDS byte offset (aperture model)
__device__ __forceinline__ AS_LDS int* lds_i32(void* p) {
    return (AS_LDS int*)(uint32_t)(uintptr_t)p;
}
__device__ __forceinline__ AS_LDS v4i* lds_v4i(void* p) {
    return (AS_LDS v4i*)(uint32_t)(uintptr_t)p;
}
__device__ __forceinline__ AS_GLOBAL int* glob_i32(const void* p) {
    return (AS_GLOBAL int*)(uintptr_t)p;
}
__device__ __forceinline__ AS_GLOBAL v4i* glob_v4i(const void* p) {
    return (AS_GLOBAL v4i*)(uintptr_t)p;
}

__device__ __forceinline__ void wait_async_zero() {
#if __has_builtin(__builtin_amdgcn_s_wait_asynccnt)
    __builtin_amdgcn_s_wait_asynccnt(0);
#elif defined(__AMDGCN__)
    asm volatile("s_wait_asynccnt 0" ::: "memory");
#endif
}

constexpr int EMBED          = 512;
constexpr int ROWS_PER_BLOCK = 8;   // 8 waves * 32 lanes = 256 threads

__global__ __launch_bounds__(256) void
embedding_gather_kernel(const int* __restrict__ y,
                        const float* __restrict__ W,
                        float* __restrict__ out,
                        int vocab, int nrows) {
    __shared__ float stage[ROWS_PER_BLOCK * EMBED];   // 16 KB of 320 KB WGP LDS

    const int wave = __builtin_amdgcn_readfirstlane((int)(threadIdx.x >> 5));
    const int lane = (int)(threadIdx.x & 31u);
    const int b    = (int)blockIdx.x * ROWS_PER_BLOCK + wave;
    if (b >= nrows) return;   // wave-uniform branch; EXEC stays all-ones per wave

    // y[b] is wave-uniform -> force scalar (s_load) path
    const int v = __builtin_amdgcn_readfirstlane(y[b]);

    float* lrow = &stage[wave * EMBED];
    float* orow = out + (size_t)b * EMBED;

#if USE_ASYNC_LDS
    // ---- async gather: 16 scattered dwords per lane, global -> LDS --------
    // global addr = [uniform scalar base, bumped by 32*vocab floats per j]
    //             + [per-lane 32-bit byte offset lane*vocab*4]  (GVS form)
    const uint32_t lane_byte = (uint32_t)lane * (uint32_t)vocab * 4u;  // < 2^31
    const char*    gbase     = (const char*)(W + v);                    // uniform
    const size_t   gstride   = (size_t)32 * (size_t)vocab * 4u;         // uniform
#pragma unroll
    for (int j = 0; j < EMBED / 32; ++j) {
        __builtin_amdgcn_global_load_async_to_lds_b32(
            glob_i32(gbase + lane_byte), lds_i32(lrow + j * 32 + lane),
            /*offset=*/0, /*cpol=*/0);
        gbase += gstride;
    }
    wait_async_zero();   // LDS writes from the async engine complete

    // ---- async coalesced store: 4 x b128 per lane, LDS -> global ----------
#pragma unroll
    for (int k = 0; k < 4; ++k) {
        const int idx = (k * 32 + lane) * 4;   // float index, 16B granularity
        __builtin_amdgcn_global_store_async_from_lds_b128(
            glob_v4i(orow + idx), lds_v4i(lrow + idx),
            /*offset=*/0, /*cpol=*/0);
    }
    wait_async_zero();
#else
    // ---- fallback: register gather + vectorized float4 stores -------------
    (void)lrow;
    const float* col = W + v;
#pragma unroll
    for (int k = 0; k < 4; ++k) {
        const int e0 = (k * 32 + lane) * 4;    // 4 consecutive e per lane
        float4 r;
        r.x = col[(size_t)(e0 + 0) * (size_t)vocab];
        r.y = col[(size_t)(e0 + 1) * (size_t)vocab];
        r.z = col[(size_t)(e0 + 2) * (size_t)vocab];
        r.w = col[(size_t)(e0 + 3) * (size_t)vocab];
        *(float4*)(orow + e0) = r;
    }
#endif
}

extern "C" void kernel_launch(void* const* d_in, const int* in_sizes, int n_in,
                              void* d_out, int out_size, void* d_ws, size_t ws_size,
                              hipStream_t stream) {
    const int*   y = (const int*)d_in[0];     // setup_inputs order: y, W
    const float* W = (const float*)d_in[1];
    float*       o = (float*)d_out;

    const int nrows = in_sizes[0];            // B = 16384
    const int vocab = in_sizes[1] / EMBED;    // 128000

    const int blocks = (nrows + ROWS_PER_BLOCK - 1) / ROWS_PER_BLOCK;
    embedding_gather_kernel<<<blocks, 256, 0, stream>>>(y, W, o, vocab, nrows);
}